// CRF_68221260529804
// MI455X (gfx1250) — compile-verified
//
#include <hip/hip_runtime.h>
#include <stdint.h>

// Viterbi CRF decode: feats [512, 8192, 10] f32, transitions [10,10] f32.
// Outputs: path_score [8192] f32, best_path [512, 8192] (written as float).
// Latency/bandwidth bound (~185 MB HBM traffic, ~0.8 G simple ops); WMMA does
// not apply (max-plus + argmax). CDNA5 features: 16-deep TDM async tensor-load
// pipeline (TENSORcnt), single-wave32 workgroups, 148KB LDS/block (2 blocks
// co-resident per 320KB WGP) holding all backpointers -> zero bp HBM traffic.

#define TAGSET    10
#define START_TAG 8
#define STOP_TAG  9
#define SEQ       512
#define BATCH     8192
#define NEG_INF   (-10000.0f)
#define BLOCK     32                      // exactly one wave32; one lane per chain
#define PIPE      16                      // TDM prefetch depth

#define BP_BYTES    (SEQ * BLOCK * 8)     // 131072 B packed backpointers
#define TILE_FLOATS (BLOCK * TAGSET)      // 320 floats per time-step tile
#define TILE_BYTES  (TILE_FLOATS * 4)     // 1280 B
#define STAGE_BASE  BP_BYTES
#define TRANS_BASE  (BP_BYTES + PIPE * TILE_BYTES)        // 151552
#define SMEM_BYTES  (TRANS_BASE + TAGSET * TAGSET * 4)    // 151952 B per block

typedef __attribute__((ext_vector_type(4))) unsigned int u32x4;
typedef __attribute__((ext_vector_type(8))) int          i32x8;
typedef __attribute__((ext_vector_type(4))) int          i32x4;

extern __shared__ char smem_raw[];

// Issue one TDM DMA: 320 contiguous f32 (this block's feats tile for one t)
// from global memory into LDS at byte offset lds_off. Tracked by TENSORcnt.
__device__ __forceinline__ void tdm_load_tile(const float* gsrc, uint32_t lds_off) {
  uint64_t ga = (uint64_t)(uintptr_t)gsrc;

  // D# group 0: count=1 | lds_addr | global_addr[56:0] | type=2 ("image")
  u32x4 g0;
  g0.x = 1u;                                                   // count=1, user mode
  g0.y = lds_off;                                              // lds_addr (bytes)
  g0.z = (uint32_t)ga;                                         // global_addr[31:0]
  g0.w = (uint32_t)((ga >> 32) & 0x01FFFFFFu) | (2u << 30);    // addr[56:32], type=2

  // D# group 1: data_size=4B, 1-D tile of 320 elements, stride 320.
  i32x8 g1;
  g1[0] = (int)(2u << 16);                                 // mask=0, data_size=2 (4B)
  g1[1] = (int)(((unsigned)TILE_FLOATS & 0xFFFFu) << 16);  // tensor_dim0 lo16 @63:48
  g1[2] = (int)(1u << 16);                                 // dim0 hi=0, tensor_dim1=1
  g1[3] = (int)(((unsigned)TILE_FLOATS & 0xFFFFu) << 16);  // tile_dim0 @127:112
  g1[4] = 0;                                               // tile_dim1=0, tile_dim2=0
  g1[5] = TILE_FLOATS;                                     // tensor_dim0_stride lo32
  g1[6] = 0;
  g1[7] = 0;

  i32x4 z4 = {0, 0, 0, 0};                                 // groups 2/3 unused (<=2D)
  i32x8 z8 = {0, 0, 0, 0, 0, 0, 0, 0};
  __builtin_amdgcn_tensor_load_to_lds(g0, g1, z4, z4, z8, 0);
}

__global__ __launch_bounds__(BLOCK)
void crf_viterbi_kernel(const float* __restrict__ feats,
                        const float* __restrict__ trans,
                        float* __restrict__ out) {
  const int tid = threadIdx.x;
  const int b   = blockIdx.x * BLOCK + tid;

  uint64_t* bp     = (uint64_t*)smem_raw;               // [SEQ][BLOCK] nibbles
  float*    fstage = (float*)(smem_raw + STAGE_BASE);   // PIPE x [BLOCK*TAGSET]
  float*    ltr    = (float*)(smem_raw + TRANS_BASE);   // [100] transitions

  // Stage transitions through LDS so they live in VGPRs (avoids the 100-SGPR
  // blowup + spills). Single wave: LDS write->read ordering handled by DScnt.
  for (int i = tid; i < TAGSET * TAGSET; i += BLOCK) ltr[i] = trans[i];

  float tr[TAGSET * TAGSET];
#pragma unroll
  for (int i = 0; i < TAGSET * TAGSET; ++i) tr[i] = ltr[i];

  // init: -10000 everywhere except START_TAG = 0 (matches reference exactly).
  float fv[TAGSET];
#pragma unroll
  for (int j = 0; j < TAGSET; ++j) fv[j] = NEG_INF;
  fv[START_TAG] = 0.0f;

  // Block's slice of feats for time t: feats + t*BATCH*TAGSET + blockIdx*BLOCK*TAGSET
  const float* gbase = feats + (size_t)blockIdx.x * BLOCK * TAGSET;

  // Prologue: fill the PIPE-deep DMA pipeline.
#pragma unroll
  for (int i = 0; i < PIPE; ++i)
    tdm_load_tile(gbase + (size_t)i * BATCH * TAGSET,
                  (uint32_t)(STAGE_BASE + i * TILE_BYTES));

  for (int t = 0; t < SEQ; ++t) {
    if (t + PIPE <= SEQ) {
      __builtin_amdgcn_s_wait_tensorcnt(PIPE - 1);   // tile t landed (in-order)
    } else {
      __builtin_amdgcn_s_wait_tensorcnt(0);          // pipeline drain (last 15 iters)
    }

    const float* frow = fstage + (t % PIPE) * TILE_FLOATS + tid * TAGSET;
    float feat[TAGSET];
#pragma unroll
    for (int k = 0; k < TAGSET; ++k) feat[k] = frow[k];

    // fv_new[next] = max_prev(fv[prev] + T[next][prev]) + feat[next];
    // strict '>' keeps the earliest max, matching jnp.argmax tie-breaking.
    float nfv[TAGSET];
    uint64_t word = 0;
#pragma unroll
    for (int nx = 0; nx < TAGSET; ++nx) {
      float best = fv[0] + tr[nx * TAGSET + 0];
      int   arg  = 0;
#pragma unroll
      for (int pv = 1; pv < TAGSET; ++pv) {
        float c  = fv[pv] + tr[nx * TAGSET + pv];
        bool  gt = c > best;
        best = gt ? c : best;
        arg  = gt ? pv : arg;
      }
      nfv[nx] = best + feat[nx];
      word |= (uint64_t)(uint32_t)arg << (4 * nx);
    }
#pragma unroll
    for (int j = 0; j < TAGSET; ++j) fv[j] = nfv[j];

    bp[(size_t)t * BLOCK + tid] = word;   // 8B/lane, conflict-free b64

    // Ensure this wave's LDS reads of buffer (t % PIPE) completed before the
    // TDM engine overwrites it with tile t+PIPE.
    asm volatile("s_wait_dscnt 0" ::: "memory");
    if (t + PIPE < SEQ)
      tdm_load_tile(gbase + (size_t)(t + PIPE) * BATCH * TAGSET,
                    (uint32_t)(STAGE_BASE + ((t + PIPE) % PIPE) * TILE_BYTES));
  }

  // Terminal: path_score / last_tag from fv + T[STOP_TAG][:].
  float best = fv[0] + tr[STOP_TAG * TAGSET + 0];
  int   tag  = 0;
#pragma unroll
  for (int j = 1; j < TAGSET; ++j) {
    float c  = fv[j] + tr[STOP_TAG * TAGSET + j];
    bool  gt = c > best;
    best = gt ? c : best;
    tag  = gt ? j : tag;
  }
  out[b] = best;                           // path_score [BATCH] first

  // Backtrack from LDS backpointers; coalesced float stores of best_path.
  float* path = out + BATCH;
  uint32_t tg = (uint32_t)tag;
  for (int t = SEQ - 1; t >= 0; --t) {
    path[(size_t)t * BATCH + b] = (float)tg;
    uint64_t w = bp[(size_t)t * BLOCK + tid];
    tg = (uint32_t)(w >> (4u * tg)) & 0xFu;
  }
}

extern "C" void kernel_launch(void* const* d_in, const int* in_sizes, int n_in,
                              void* d_out, int out_size, void* d_ws, size_t ws_size,
                              hipStream_t stream) {
  (void)in_sizes; (void)n_in; (void)d_ws; (void)ws_size; (void)out_size;
  const float* feats = (const float*)d_in[0];   // [512, 8192, 10] f32
  const float* trans = (const float*)d_in[1];   // [10, 10] f32
  float* out = (float*)d_out;                   // [8192] + [512*8192]

  dim3 grid(BATCH / BLOCK);   // 256 single-wave blocks; 2 co-resident per WGP
  dim3 block(BLOCK);          // 32 threads = 1 wave32
  hipLaunchKernelGGL(crf_viterbi_kernel, grid, block, SMEM_BYTES, stream,
                     feats, trans, out);
}